// MoAT_79293686219467
// MI455X (gfx1250) — compile-verified
//
#include <hip/hip_runtime.h>
#include <math.h>

// ---------------------------------------------------------------------------
// MoAT forward for MI455X (gfx1250, wave32).
// Pipeline:
//   k_vsoftmax : V = softmax(V_compress, axis=1)                (512x2)
//   k_ipf      : per-(i,j) 2x2 IPF, 10 iters, mask diag, clip   (512x512x2x2)
//   k_wm       : Wm = sym(tril(sigmoid(W),-1))                  (512x512)
//   k_build_l0 : shifted/padded L0[1:,1:] -> mats slot 128
//   k_build_lb : shifted/padded Lb[b,1:,1:] -> mats slots 0..127
//   k_lu_logdet: blocked unpivoted LU, trailing update via
//                V_WMMA_F32_16X16X4_F32 (1x4 tile register blocking)
//   k_final    : out[b] = sum_i log Pr[b,i] + ld[b] - ld[128]
// ---------------------------------------------------------------------------

#define NN    512
#define BATCH 128
#define NMAT  129          // 128 batch Laplacians + L0
#define NB    16
#define NIPF  10
#define EPSC  1e-07f

typedef float v2f __attribute__((ext_vector_type(2)));
typedef float v8f __attribute__((ext_vector_type(8)));

// ---------------- V = softmax(V_compress, axis=1) ----------------
__global__ void k_vsoftmax(const float* __restrict__ Vc, float* __restrict__ V) {
  int i = blockIdx.x * blockDim.x + threadIdx.x;
  if (i >= NN) return;
  float v0 = Vc[2 * i], v1 = Vc[2 * i + 1];
  float m  = fmaxf(v0, v1);
  float e0 = expf(v0 - m), e1 = expf(v1 - m);
  float inv = 1.0f / (e0 + e1);
  V[2 * i]     = e0 * inv;
  V[2 * i + 1] = e1 * inv;
}

// ---------------- per-(i,j) 2x2 IPF ----------------
__global__ void k_ipf(const float* __restrict__ Ec, const float* __restrict__ V,
                      float* __restrict__ E) {
  int idx = blockIdx.x * blockDim.x + threadIdx.x;
  if (idx >= NN * NN) return;
  int i = idx >> 9, j = idx & (NN - 1);

  float4 ein = ((const float4*)Ec)[idx];
  float e00 = expf(ein.x), e01 = expf(ein.y), e10 = expf(ein.z), e11 = expf(ein.w);
  float inv = 1.0f / (e00 + e01 + e10 + e11);
  e00 *= inv; e01 *= inv; e10 *= inv; e11 *= inv;

  float a0 = V[2 * i], a1 = V[2 * i + 1];
  float b0 = V[2 * j], b1 = V[2 * j + 1];

#pragma unroll
  for (int it = 0; it < NIPF; ++it) {
    float r0 = e00 + e01 + EPSC, r1 = e10 + e11 + EPSC;
    float c0 = e00 + e10 + EPSC, c1 = e01 + e11 + EPSC;
    float fa0 = a0 / r0, fa1 = a1 / r1;
    float fb0 = b0 / c0, fb1 = b1 / c1;
    e00 *= fa0 * fb0; e01 *= fa0 * fb1;
    e10 *= fa1 * fb0; e11 *= fa1 * fb1;
    float s = e00 + e01 + e10 + e11 + EPSC;
    float is = 1.0f / s;
    e00 *= is; e01 *= is; e10 *= is; e11 *= is;
  }
  if (i == j) { e00 = e01 = e10 = e11 = 0.0f; }
  e00 = fminf(fmaxf(e00, 0.0f), 1.0f);
  e01 = fminf(fmaxf(e01, 0.0f), 1.0f);
  e10 = fminf(fmaxf(e10, 0.0f), 1.0f);
  e11 = fminf(fmaxf(e11, 0.0f), 1.0f);
  ((float4*)E)[idx] = make_float4(e00, e01, e10, e11);
}

// ---------------- Wm = sym(tril(sigmoid(W),-1)) ----------------
__global__ void k_wm(const float* __restrict__ W, float* __restrict__ Wm) {
  int idx = blockIdx.x * blockDim.x + threadIdx.x;
  if (idx >= NN * NN) return;
  int i = idx >> 9, j = idx & (NN - 1);
  float w = (i > j) ? W[i * NN + j] : W[j * NN + i];
  float s = 1.0f / (1.0f + expf(-w));
  Wm[idx] = (i == j) ? 0.0f : s;
}

// ---------------- build shifted/padded L0 into slot 128 ----------------
__global__ void k_build_l0(const float* __restrict__ Wm, float* __restrict__ Mg) {
  int r = blockIdx.x;                 // row of the shifted 512x512 matrix
  int tid = threadIdx.x;
  float* row = Mg + (size_t)r * NN;
  if (r == NN - 1) {                  // padding row -> identity
    for (int c = tid; c < NN; c += 256) row[c] = (c == NN - 1) ? 1.0f : 0.0f;
    return;
  }
  int i = r + 1;
  __shared__ float red[256];
  float s = 0.0f;
  for (int j = tid; j < NN; j += 256) s += Wm[i * NN + j];
  red[tid] = s;
  __syncthreads();
  for (int off = 128; off > 0; off >>= 1) {
    if (tid < off) red[tid] += red[tid + off];
    __syncthreads();
  }
  float rowsum = red[0];
  for (int c = tid; c < NN; c += 256) {
    float val;
    if (c == NN - 1) val = 0.0f;
    else { int j = c + 1; val = (j == i) ? rowsum : -Wm[i * NN + j]; }
    row[c] = val;
  }
}

// ---------------- build shifted/padded Lb[b] into slots 0..127 ----------------
__global__ void k_build_lb(const float* __restrict__ Wm, const float* __restrict__ E,
                           const float* __restrict__ V, const int* __restrict__ x,
                           float* __restrict__ mats) {
  int r = blockIdx.x;
  int b = blockIdx.y;
  int tid = threadIdx.x;
  float* row = mats + (size_t)b * NN * NN + (size_t)r * NN;
  if (r == NN - 1) {
    for (int c = tid; c < NN; c += 256) row[c] = (c == NN - 1) ? 1.0f : 0.0f;
    return;
  }
  int i  = r + 1;
  int xi = x[b * NN + i];
  float pri = V[2 * i + xi];

  __shared__ float wp[NN];
  __shared__ float red[256];
  float s = 0.0f;
  for (int j = tid; j < NN; j += 256) {
    int xj = x[b * NN + j];
    float prj = V[2 * j + xj];
    float e = E[(((size_t)i * NN + j) << 2) + (xi * 2 + xj)];
    float w = Wm[i * NN + j] * (e / (pri * prj));
    wp[j] = w;
    s += w;
  }
  red[tid] = s;
  __syncthreads();
  for (int off = 128; off > 0; off >>= 1) {
    if (tid < off) red[tid] += red[tid + off];
    __syncthreads();
  }
  float rowsum = red[0];
  for (int c = tid; c < NN; c += 256) {
    float val;
    if (c == NN - 1) val = 0.0f;
    else { int j = c + 1; val = (j == i) ? rowsum : -wp[j]; }
    row[c] = val;
  }
}

// ---------------- blocked unpivoted LU + logdet, WMMA trailing update ------
// One workgroup (8 wave32) per matrix. Panels solved scalar (O(N^2)),
// Schur complement C -= L21*U12 done with V_WMMA_F32_16X16X4_F32.
// L21 is stored NEGATED in LDS (F32 WMMA has no A/B negate modifier).
// Trailing GEMM register-blocks 1x4 tiles per wave: one A-fragment feeds
// four WMMAs (16x64 strip of C), cutting LDS A-traffic 4x.
__global__ void __launch_bounds__(256)
k_lu_logdet(float* __restrict__ mats, float* __restrict__ ld) {
  int g = blockIdx.x;
  float* Mg = mats + (size_t)g * NN * NN;

  __shared__ float sD[NB * (NB + 1)];       // diagonal block (padded rows)
  __shared__ float sL[(NN - NB) * NB];      // negated L21 panel [rowLocal][k]
  __shared__ float sAcc;

  int tid  = threadIdx.x;
  int lane = tid & 31;
  int wave = tid >> 5;
  int half = lane >> 4;                     // 0: lanes 0-15, 1: lanes 16-31
  int l16  = lane & 15;

  if (tid == 0) sAcc = 0.0f;
  __syncthreads();

  for (int kb = 0; kb < NN; kb += NB) {
    int rem = NN - kb - NB;

    // ---- load 16x16 diagonal block into LDS
    {
      int r = tid >> 4, c = tid & 15;       // 256 threads == 256 elements
      sD[r * (NB + 1) + c] = Mg[(size_t)(kb + r) * NN + kb + c];
    }
    __syncthreads();

    // ---- tiny serial LU of the diagonal block + logdet accumulation
    if (tid == 0) {
      float acc = 0.0f;
      for (int p = 0; p < NB; ++p) {
        float piv = sD[p * (NB + 1) + p];
        acc += logf(fabsf(piv));
        float inv = 1.0f / piv;
        for (int r = p + 1; r < NB; ++r) {
          float l = sD[r * (NB + 1) + p] * inv;
          sD[r * (NB + 1) + p] = l;
          for (int c = p + 1; c < NB; ++c)
            sD[r * (NB + 1) + c] -= l * sD[p * (NB + 1) + c];
        }
      }
      sAcc += acc;
    }
    __syncthreads();

    if (rem > 0) {
      // ---- L21 = A21 * U11^-1  (per-row forward solve), stored negated in LDS
      for (int rr = tid; rr < rem; rr += 256) {
        int r = kb + NB + rr;
        const float* src = Mg + (size_t)r * NN + kb;
        float v[NB];
#pragma unroll
        for (int c = 0; c < NB; ++c) v[c] = src[c];
#pragma unroll
        for (int c = 0; c < NB; ++c) {
          float t = v[c];
          for (int p = 0; p < c; ++p) t -= v[p] * sD[p * (NB + 1) + c];
          t /= sD[c * (NB + 1) + c];
          v[c] = t;
          sL[rr * NB + c] = -t;
        }
      }
      // ---- U12 = L11^-1 * A12 (per-column solve), written back in place
      for (int cc = tid; cc < rem; cc += 256) {
        int c = kb + NB + cc;
        float v[NB];
#pragma unroll
        for (int r = 0; r < NB; ++r) v[r] = Mg[(size_t)(kb + r) * NN + c];
#pragma unroll
        for (int r = 1; r < NB; ++r) {
          float t = v[r];
          for (int p = 0; p < r; ++p) t -= sD[r * (NB + 1) + p] * v[p];
          v[r] = t;
        }
#pragma unroll
        for (int r = 0; r < NB; ++r) Mg[(size_t)(kb + r) * NN + c] = v[r];
      }
      __threadfence();
      __syncthreads();

      // ---- trailing update A22 += (-L21) * U12 via f32 WMMA
      int nt     = rem >> 4;                // 16x16 tiles per side
      int ntFull = nt & ~3;                 // tile-columns handled in 1x4 strips
      int nsf    = ntFull >> 2;             // strips per tile-row
      int nStrip = nt * nsf;                // strip work items
      int nRemC  = nt - ntFull;             // leftover tile-columns (0..3)
      int nWork  = nStrip + nt * nRemC;

      for (int t = wave; t < nWork; t += 8) {     // wave-uniform loop
        if (t < nStrip) {
          // ---- fast path: 16x64 strip (4 C tiles share one A fragment)
          int ti  = t / nsf, sj = t - ti * nsf;
          int rt  = kb + NB + ti * NB;
          int ct0 = kb + NB + sj * 64;
          int arow = ti * NB + l16;

          v8f c0, c1, c2, c3;
#pragma unroll
          for (int r = 0; r < 8; ++r) {
            size_t ro = (size_t)(rt + half * 8 + r) * NN + l16;
            c0[r] = Mg[ro + ct0];
            c1[r] = Mg[ro + ct0 + 16];
            c2[r] = Mg[ro + ct0 + 32];
            c3[r] = Mg[ro + ct0 + 48];
          }
#pragma unroll
          for (int kk = 0; kk < NB; kk += 4) {
            v2f a;
            a[0] = sL[arow * NB + kk + half * 2 + 0];
            a[1] = sL[arow * NB + kk + half * 2 + 1];
            size_t b0o = (size_t)(kb + kk + half * 2 + 0) * NN + l16;
            size_t b1o = (size_t)(kb + kk + half * 2 + 1) * NN + l16;
            v2f b;
            b[0] = Mg[b0o + ct0];      b[1] = Mg[b1o + ct0];
            c0 = __builtin_amdgcn_wmma_f32_16x16x4_f32(false, a, false, b,
                                                       (short)0, c0, false, false);
            b[0] = Mg[b0o + ct0 + 16]; b[1] = Mg[b1o + ct0 + 16];
            c1 = __builtin_amdgcn_wmma_f32_16x16x4_f32(false, a, false, b,
                                                       (short)0, c1, false, false);
            b[0] = Mg[b0o + ct0 + 32]; b[1] = Mg[b1o + ct0 + 32];
            c2 = __builtin_amdgcn_wmma_f32_16x16x4_f32(false, a, false, b,
                                                       (short)0, c2, false, false);
            b[0] = Mg[b0o + ct0 + 48]; b[1] = Mg[b1o + ct0 + 48];
            c3 = __builtin_amdgcn_wmma_f32_16x16x4_f32(false, a, false, b,
                                                       (short)0, c3, false, false);
          }
#pragma unroll
          for (int r = 0; r < 8; ++r) {
            size_t ro = (size_t)(rt + half * 8 + r) * NN + l16;
            Mg[ro + ct0]      = c0[r];
            Mg[ro + ct0 + 16] = c1[r];
            Mg[ro + ct0 + 32] = c2[r];
            Mg[ro + ct0 + 48] = c3[r];
          }
        } else {
          // ---- remainder path: single 16x16 tile
          int u  = t - nStrip;
          int ti = u / nRemC;
          int tj = ntFull + (u - ti * nRemC);
          int rt = kb + NB + ti * NB;
          int ct = kb + NB + tj * NB;
          int arow = ti * NB + l16;

          v8f c;
#pragma unroll
          for (int r = 0; r < 8; ++r)
            c[r] = Mg[(size_t)(rt + half * 8 + r) * NN + ct + l16];
#pragma unroll
          for (int kk = 0; kk < NB; kk += 4) {
            v2f a, b;
            a[0] = sL[arow * NB + kk + half * 2 + 0];
            a[1] = sL[arow * NB + kk + half * 2 + 1];
            b[0] = Mg[(size_t)(kb + kk + half * 2 + 0) * NN + ct + l16];
            b[1] = Mg[(size_t)(kb + kk + half * 2 + 1) * NN + ct + l16];
            c = __builtin_amdgcn_wmma_f32_16x16x4_f32(false, a, false, b,
                                                      (short)0, c, false, false);
          }
#pragma unroll
          for (int r = 0; r < 8; ++r)
            Mg[(size_t)(rt + half * 8 + r) * NN + ct + l16] = c[r];
        }
      }
      __threadfence();
    }
    __syncthreads();
  }

  if (tid == 0) ld[g] = sAcc;
}

// ---------------- finalize: out[b] = sum log Pr + ld[b] - ld[128] ----------
__global__ void k_final(const int* __restrict__ x, const float* __restrict__ V,
                        const float* __restrict__ ld, float* __restrict__ out) {
  int b = blockIdx.x;
  int tid = threadIdx.x;
  __shared__ float red[256];
  float s = 0.0f;
  for (int i = tid; i < NN; i += 256) {
    int xi = x[b * NN + i];
    s += logf(V[2 * i + xi]);
  }
  red[tid] = s;
  __syncthreads();
  for (int off = 128; off > 0; off >>= 1) {
    if (tid < off) red[tid] += red[tid + off];
    __syncthreads();
  }
  if (tid == 0) out[b] = red[0] + ld[b] - ld[NMAT - 1];
}

// ---------------------------------------------------------------------------
extern "C" void kernel_launch(void* const* d_in, const int* in_sizes, int n_in,
                              void* d_out, int out_size, void* d_ws, size_t ws_size,
                              hipStream_t stream) {
  (void)in_sizes; (void)n_in; (void)out_size; (void)ws_size;
  const int*   x  = (const int*)  d_in[0];   // (128, 512) int32
  const float* W  = (const float*)d_in[1];   // (512, 512)
  const float* Vc = (const float*)d_in[2];   // (512, 2)
  const float* Ec = (const float*)d_in[3];   // (512, 512, 2, 2)
  float* out = (float*)d_out;                // (128,)

  // workspace layout (bytes); total requirement ~141 MB
  char* ws = (char*)d_ws;
  float* V    = (float*)(ws + 0);                              //   4 KB
  float* E    = (float*)(ws + 4096);                           //   4 MB
  float* Wm   = (float*)(ws + 4096 + 4u * 1024 * 1024);        //   1 MB
  float* ld   = (float*)(ws + 5246976u);                       // 129 f32
  float* mats = (float*)(ws + 5248000u);                       // 129 MB

  k_vsoftmax<<<2, 256, 0, stream>>>(Vc, V);
  k_ipf<<<(NN * NN) / 256, 256, 0, stream>>>(Ec, V, E);
  k_wm<<<(NN * NN) / 256, 256, 0, stream>>>(W, Wm);
  k_build_l0<<<NN, 256, 0, stream>>>(Wm, mats + (size_t)(NMAT - 1) * NN * NN);
  k_build_lb<<<dim3(NN, BATCH), 256, 0, stream>>>(Wm, E, V, x, mats);
  k_lu_logdet<<<NMAT, 256, 0, stream>>>(mats, ld);
  k_final<<<BATCH, 256, 0, stream>>>(x, V, ld, out);
}